// VQEmbeddingEMA_31482110280341
// MI455X (gfx1250) — compile-verified
//
#include <hip/hip_runtime.h>

typedef __attribute__((ext_vector_type(2))) float v2f;
typedef __attribute__((ext_vector_type(4))) float v4f;
typedef __attribute__((ext_vector_type(8))) float v8f;
typedef __attribute__((ext_vector_type(4))) int v4i;

#define VQ_B 64
#define VQ_T 2048
#define VQ_D 128
#define VQ_M 1024
#define VQ_N (VQ_B * VQ_T) /* 131072 */
#define VQ_NTILES (VQ_M / 16) /* 64 */

#define DECAY_F 0.999f
#define OMD_F 0.001f
#define EPS_F 1e-5f
#define COMMIT_F 0.25f

#define LDSB_STRIDE 132 /* 128 + 4 pad floats: kills 16-way bank conflicts */
#define LDSB_BUFSZ (16 * LDSB_STRIDE)

// ---- CDNA5 async LDS-copy path (guarded; falls back to sync staging) ------
#if defined(__AMDGCN__) && __has_builtin(__builtin_amdgcn_global_load_async_to_lds_b128)
#define VQ_ASYNC 1
#else
#define VQ_ASYNC 0
#endif

#if VQ_ASYNC
__device__ __forceinline__ void vq_async_b128(const float* g, float* l) {
  __builtin_amdgcn_global_load_async_to_lds_b128(
      (__attribute__((address_space(1))) v4i*)g,
      (__attribute__((address_space(3))) v4i*)l, 0, 0);
}
__device__ __forceinline__ void vq_wait_async_2() {
#if __has_builtin(__builtin_amdgcn_s_wait_asynccnt)
  __builtin_amdgcn_s_wait_asynccnt(2);
#else
  asm volatile("s_wait_asynccnt 0x2" ::: "memory");
#endif
}
__device__ __forceinline__ void vq_wait_async_0() {
#if __has_builtin(__builtin_amdgcn_s_wait_asynccnt)
  __builtin_amdgcn_s_wait_asynccnt(0);
#else
  asm volatile("s_wait_asynccnt 0x0" ::: "memory");
#endif
}
#endif

// ---------------------------------------------------------------------------
// Kernel 1: codeword norms + zero the accumulation buffers (every call).
// ---------------------------------------------------------------------------
__global__ void vq_prep_kernel(const float* __restrict__ emb,
                               float* __restrict__ enorm,
                               float* __restrict__ counts,
                               float* __restrict__ dw,
                               float* __restrict__ loss_acc) {
  const int gid = blockIdx.x * blockDim.x + threadIdx.x;
  const int stride = gridDim.x * blockDim.x;
  for (int i = gid; i < VQ_M * VQ_D; i += stride) dw[i] = 0.0f;
  if (gid < VQ_M) {
    counts[gid] = 0.0f;
    const float* r = emb + (size_t)gid * VQ_D;
    float s = 0.0f;
#pragma unroll 4
    for (int d = 0; d < VQ_D; ++d) s += r[d] * r[d];
    enorm[gid] = s;
  }
  if (gid == 0) loss_acc[0] = 0.0f;
}

// ---------------------------------------------------------------------------
// Kernel 2: fused fp32-WMMA distance GEMM + argmin + gather + scatter.
// Block = 256 threads = 8 waves; each wave owns 16 rows of x; block loops
// over all 64 codebook column-tiles, double-buffered through LDS via
// GLOBAL_LOAD_ASYNC_TO_LDS_B128 when available.
// ---------------------------------------------------------------------------
__global__ __launch_bounds__(256) void vq_assign_kernel(
    const float* __restrict__ x, const float* __restrict__ emb,
    const float* __restrict__ enorm,
    float* __restrict__ q_out, float* __restrict__ codes_out,
    float* __restrict__ counts, float* __restrict__ dw,
    float* __restrict__ loss_acc) {
#if VQ_ASYNC
  __shared__ float ldsB[2 * LDSB_BUFSZ];
#else
  __shared__ float ldsB[LDSB_BUFSZ];
#endif

  const int lane = threadIdx.x & 31;
  const int wave = threadIdx.x >> 5;
  const int rowbase = blockIdx.x * 128 + wave * 16;

  // Cooperative B-tile staging assignment: thread -> (codeword row, 8 floats).
  const int bc = threadIdx.x >> 4;        // 0..15 codeword within tile
  const int bk = (threadIdx.x & 15) * 8;  // 0..120

  // A fragment layout (V_WMMA_F32_16X16X4_F32): lanes 0-15 rows 0-15 with
  // K={4k,4k+1}; lanes 16-31 same rows with K={4k+2,4k+3}.
  const int arow = rowbase + (lane & 15);
  const int koff = (lane >> 4) << 1; // 0 or 2
  const float* xrow = x + (size_t)arow * VQ_D;
  v2f afrag[32];
#pragma unroll
  for (int ks = 0; ks < 32; ++ks)
    afrag[ks] = *(const v2f*)(xrow + ks * 4 + koff);

  float minv[8];
  unsigned mini[8];
#pragma unroll
  for (int v = 0; v < 8; ++v) { minv[v] = 3.4e38f; mini[v] = 0u; }

  const int ccol = lane & 15;

#if VQ_ASYNC
  // Prefetch tile 0 into buffer 0 (2 x b128 per thread).
  {
    const float* src = emb + (size_t)bc * VQ_D + bk;
    float* dst = &ldsB[bc * LDSB_STRIDE + bk];
    vq_async_b128(src, dst);
    vq_async_b128(src + 4, dst + 4);
  }
#endif

  for (int t = 0; t < VQ_NTILES; ++t) {
#if VQ_ASYNC
    if (t + 1 < VQ_NTILES) {
      // Prefetch next tile into the other buffer, then wait for current tile.
      const float* src = emb + (size_t)((t + 1) * 16 + bc) * VQ_D + bk;
      float* dst = &ldsB[((t + 1) & 1) * LDSB_BUFSZ + bc * LDSB_STRIDE + bk];
      vq_async_b128(src, dst);
      vq_async_b128(src + 4, dst + 4);
      vq_wait_async_2();
    } else {
      vq_wait_async_0();
    }
    __syncthreads(); // all waves' async writes for tile t are in LDS
    const float* bbase = &ldsB[(t & 1) * LDSB_BUFSZ];
#else
    __syncthreads();
    {
      const float* src = emb + (size_t)(t * 16 + bc) * VQ_D + bk;
      float* dst = &ldsB[bc * LDSB_STRIDE + bk];
#pragma unroll
      for (int j = 0; j < 8; j += 4) *(v4f*)(dst + j) = *(const v4f*)(src + j);
    }
    __syncthreads();
    const float* bbase = ldsB;
#endif

    v8f cacc = {0.f, 0.f, 0.f, 0.f, 0.f, 0.f, 0.f, 0.f};
#pragma unroll
    for (int ks = 0; ks < 32; ++ks) {
      // B fragment: lane holds B[k][col=ccol] for k = 4*ks + koff + {0,1},
      // i.e. embedding[col][k..k+1] (LDS row-major, padded stride).
      v2f bfrag = *(const v2f*)(bbase + ccol * LDSB_STRIDE + ks * 4 + koff);
      cacc = __builtin_amdgcn_wmma_f32_16x16x4_f32(
          false, afrag[ks], false, bfrag, (short)0, cacc, false, false);
    }

    // dist = ||e||^2 - 2 x.e  (||x||^2 is row-constant, irrelevant to argmin)
    const float en = enorm[t * 16 + ccol];
    const unsigned col = (unsigned)(t * 16 + ccol);
#pragma unroll
    for (int v = 0; v < 8; ++v) {
      const float dist = en - 2.0f * cacc[v];
      if (dist < minv[v]) { minv[v] = dist; mini[v] = col; }
    }

#if VQ_ASYNC
    __syncthreads(); // buffer (t&1) free for reuse before next prefetch
#endif
  }

  // Cross-lane argmin: slot v holds row v (lanes 0-15) / row v+8 (lanes 16-31);
  // butterfly within each 16-lane half, ties -> lowest index (matches jnp.argmin).
#pragma unroll
  for (int v = 0; v < 8; ++v) {
    float val = minv[v];
    unsigned idx = mini[v];
#pragma unroll
    for (int off = 8; off >= 1; off >>= 1) {
      const float ov = __shfl_xor(val, off, 32);
      const unsigned oi = (unsigned)__shfl_xor((int)idx, off, 32);
      if (ov < val || (ov == val && oi < idx)) { val = ov; idx = oi; }
    }
    minv[v] = val;
    mini[v] = idx;
  }

  // Gather winners, write quantized (== quantized_st numerically), codes,
  // commitment-loss partial, and EMA scatter (dw, counts).
  float lsum = 0.0f;
#pragma unroll
  for (int r = 0; r < 16; ++r) {
    const int srcLane = (r >> 3) << 4; // 0 or 16
    const unsigned idx = (unsigned)__shfl((int)mini[r & 7], srcLane, 32);
    const int row = rowbase + r;
    const float* er = emb + (size_t)idx * VQ_D + lane * 4;
    const float* xr = x + (size_t)row * VQ_D + lane * 4;
    const v4f e = *(const v4f*)er;
    const v4f xv = *(const v4f*)xr;
    *(v4f*)(q_out + (size_t)row * VQ_D + lane * 4) = e;
#pragma unroll
    for (int j = 0; j < 4; ++j) {
      const float d = xv[j] - e[j];
      lsum += d * d;
      atomicAdd(&dw[(size_t)idx * VQ_D + lane * 4 + j], xv[j]);
    }
    if (lane == 0) {
      codes_out[row] = (float)idx;
      atomicAdd(&counts[idx], 1.0f);
    }
  }

#pragma unroll
  for (int off = 16; off >= 1; off >>= 1) lsum += __shfl_xor(lsum, off, 32);
  if (lane == 0) atomicAdd(loss_acc, lsum);
}

// ---------------------------------------------------------------------------
// Kernel 3: EMA count update + normalization, perplexity, loss scalar.
// One block of 1024 threads (one per codeword).
// ---------------------------------------------------------------------------
__global__ __launch_bounds__(1024) void vq_finalize_kernel(
    const float* __restrict__ ema_count, const float* __restrict__ counts,
    const float* __restrict__ loss_acc, float* __restrict__ out_loss,
    float* __restrict__ out_perp, float* __restrict__ out_nec) {
  __shared__ float redA[1024];
  __shared__ float redB[1024];
  const int m = threadIdx.x;
  const float c = counts[m];
  const float nec_raw = DECAY_F * ema_count[m] + OMD_F * c;
  const float p = c * (1.0f / (float)VQ_N);
  redA[m] = nec_raw;
  redB[m] = -p * logf(p + 1e-10f);
  __syncthreads();
  for (int s = 512; s > 0; s >>= 1) {
    if (m < s) {
      redA[m] += redA[m + s];
      redB[m] += redB[m + s];
    }
    __syncthreads();
  }
  const float n = redA[0];
  out_nec[m] = (nec_raw + EPS_F) / (n + (float)VQ_M * EPS_F) * n;
  if (m == 0) {
    out_perp[0] = expf(redB[0]);
    out_loss[0] = COMMIT_F * loss_acc[0] * (1.0f / (float)((size_t)VQ_N * VQ_D));
  }
}

// ---------------------------------------------------------------------------
// Kernel 4: EMA weight update + new embedding.
// ---------------------------------------------------------------------------
__global__ void vq_weights_kernel(const float* __restrict__ ema_weight,
                                  const float* __restrict__ dw,
                                  const float* __restrict__ nec,
                                  float* __restrict__ out_emb,
                                  float* __restrict__ out_w) {
  const int i = blockIdx.x * blockDim.x + threadIdx.x;
  if (i < VQ_M * VQ_D) {
    const float w = DECAY_F * ema_weight[i] + OMD_F * dw[i];
    out_w[i] = w;
    out_emb[i] = w / nec[i >> 7]; // D = 128
  }
}

// ---------------------------------------------------------------------------
extern "C" void kernel_launch(void* const* d_in, const int* in_sizes, int n_in,
                              void* d_out, int out_size, void* d_ws,
                              size_t ws_size, hipStream_t stream) {
  const float* x = (const float*)d_in[0];          // [64,2048,128]
  const float* emb = (const float*)d_in[1];        // [1024,128]
  const float* ema_count = (const float*)d_in[2];  // [1024]
  const float* ema_weight = (const float*)d_in[3]; // [1024,128]

  // Output tuple, flattened in return order.
  float* out = (float*)d_out;
  float* q_out = out;                                 // 16777216
  float* codes = q_out + (size_t)VQ_N * VQ_D;         // 131072
  float* loss = codes + VQ_N;                         // 1
  float* perp = loss + 1;                             // 1
  float* new_emb = perp + 1;                          // 131072
  float* new_cnt = new_emb + (size_t)VQ_M * VQ_D;     // 1024
  float* new_w = new_cnt + VQ_M;                      // 131072

  // Workspace: enorm[1024] | counts[1024] | dw[1024*128] | loss_acc[1]
  float* ws = (float*)d_ws;
  float* enorm = ws;
  float* counts = ws + VQ_M;
  float* dw = ws + 2 * VQ_M;
  float* lossacc = dw + (size_t)VQ_M * VQ_D;

  vq_prep_kernel<<<512, 256, 0, stream>>>(emb, enorm, counts, dw, lossacc);
  vq_assign_kernel<<<VQ_N / 128, 256, 0, stream>>>(x, emb, enorm, q_out, codes,
                                                   counts, dw, lossacc);
  vq_finalize_kernel<<<1, 1024, 0, stream>>>(ema_count, counts, lossacc, loss,
                                             perp, new_cnt);
  vq_weights_kernel<<<(VQ_M * VQ_D + 255) / 256, 256, 0, stream>>>(
      ema_weight, dw, new_cnt, new_emb, new_w);
}